// Cosal_Module_19533511262862
// MI455X (gfx1250) — compile-verified
//
#include <hip/hip_runtime.h>
#include <math.h>

#define NIMG 16
#define CCH  256
#define HWPX 4096
#define CHW  (CCH * HWPX)          // 1048576
#define NPIX (NIMG * HWPX)         // 65536
#define EPSF 1e-12f

typedef float v2f __attribute__((ext_vector_type(2)));
typedef float v8f __attribute__((ext_vector_type(8)));

#define WMMA_F32(a, b, c) \
  __builtin_amdgcn_wmma_f32_16x16x4_f32(false, (a), false, (b), (short)0, (c), false, false)

// ---------------- workspace layout (float offsets) ----------------
#define WS_SIG    0                               // [16][4096] sigmoid(SISM)
#define WS_INVN   (WS_SIG + NPIX)                 // [16][4096] 1/max(||feat||,eps)
#define WS_WEI    (WS_INVN + NPIX)                // [2][16][256] weighted means
#define WS_PROXY  (WS_WEI + 2*NIMG*CCH)           // [2][16][256] co_proxy
#define WS_RPROXY (WS_PROXY + 2*NIMG*CCH)         // [2][256]     r_co_proxy
#define WS_RN     (WS_RPROXY + 2*CCH)             // [2][16][16]  1/corr row norms
#define WS_ATT    (WS_RN + 2*NIMG*NIMG)           // [2][16][4096] normalized attention
#define WS_SCORE  (WS_ATT + 2*NIMG*HWPX)          // [2][65536]
#define WS_COREP  (WS_SCORE + 2*NPIX)             // [2][32][256]
#define WS_IDX    (WS_COREP + 2*32*CCH)           // [2][32] (ints)
#define WS_CORR   (WS_IDX + 64)                   // [2][16][16][4096]
#define WS_TATT   (WS_CORR + 2*NIMG*NIMG*HWPX)    // [2][16][32][4096] CoFs*att

// =============== small prep kernels ===============

__global__ void prep_kernel(const float* __restrict__ feats, const float* __restrict__ sisms,
                            float* __restrict__ sig, float* __restrict__ invn) {
  int g = blockIdx.x * 256 + threadIdx.x;            // 0..65535
  sig[g] = 1.0f / (1.0f + expf(-sisms[g]));
  int n = g >> 12, p = g & (HWPX - 1);
  const float* f = feats + (size_t)n * CHW + p;
  float s = 0.0f;
  for (int c = 0; c < CCH; ++c) { float v = f[(size_t)c * HWPX]; s += v * v; }
  invn[g] = 1.0f / fmaxf(sqrtf(s), EPSF);
}

// weighted[br][n][c] = mean_p NFs * S_br  (back = all - fore)
__global__ void weighted_kernel(const float* __restrict__ feats, const float* __restrict__ sig,
                                const float* __restrict__ invn, float* __restrict__ wei) {
  int n = blockIdx.x >> 8, c = blockIdx.x & 255, tid = threadIdx.x;
  const float* f  = feats + (size_t)n * CHW + (size_t)c * HWPX;
  const float* sg = sig  + n * HWPX;
  const float* iv = invn + n * HWPX;
  float sa = 0.0f, sf = 0.0f;
  for (int p = tid; p < HWPX; p += 256) { float v = f[p] * iv[p]; sa += v; sf += v * sg[p]; }
  __shared__ float ra[256], rf[256];
  ra[tid] = sa; rf[tid] = sf; __syncthreads();
  for (int off = 128; off; off >>= 1) {
    if (tid < off) { ra[tid] += ra[tid + off]; rf[tid] += rf[tid + off]; }
    __syncthreads();
  }
  if (tid == 0) {
    wei[(0 * NIMG + n) * CCH + c] = rf[0] * (1.0f / HWPX);
    wei[(1 * NIMG + n) * CCH + c] = (ra[0] - rf[0]) * (1.0f / HWPX);
  }
}

// co_proxy (rows 0..15) and r_co_proxy (row 16); grid (17,2) x 256
__global__ void proxy_kernel(const float* __restrict__ wei, float* __restrict__ proxy,
                             float* __restrict__ rproxy) {
  int i = blockIdx.x, br = blockIdx.y, tid = threadIdx.x;
  float v;
  if (i < NIMG) v = wei[(br * NIMG + i) * CCH + tid];
  else { v = 0.0f; for (int n = 0; n < NIMG; ++n) v += wei[(br * NIMG + n) * CCH + tid];
         v *= (1.0f / NIMG); }
  __shared__ float red[256];
  red[tid] = v * v; __syncthreads();
  for (int off = 128; off; off >>= 1) {
    if (tid < off) red[tid] += red[tid + off];
    __syncthreads();
  }
  float nrm = fmaxf(sqrtf(red[0]), EPSF);
  if (i < NIMG) proxy[(br * NIMG + i) * CCH + tid] = v / nrm;
  else          rproxy[br * CCH + tid] = v / nrm;
}

// =============== WMMA GEMM kernels (f32 16x16x4) ===============
// A-frag lane l, half=lane>>4: (A[m=l, kb], A[m=l, kb+1]), kb = k + 2*half
// B-frag: (B[kb, j=l], B[kb+1, j=l]); C-frag VGPR r -> row r+8*half, col lane&15.

// corr[br][n][m][p] = invn[n][p] * sum_c proxy[br][m][c]*feats[n][c][p]
// wave: 16 rows x 64 pixels (4 p-tiles share the A fragment); grid (16,16,2) x 128
__global__ void corr_gemm(const float* __restrict__ feats, const float* __restrict__ invn,
                          const float* __restrict__ proxy, float* __restrict__ corr) {
  const int lane = threadIdx.x & 31, wid = threadIdx.x >> 5;
  const int half = lane >> 4, l = lane & 15;
  const int p0 = (blockIdx.x * 4 + wid) << 6;                // 4 pixel tiles
  const int n = blockIdx.y, br = blockIdx.z;
  const float* A = proxy + br * NIMG * CCH;                  // [16][256]
  const float* B = feats + (size_t)n * CHW + p0;
  v8f acc[4] = {};
#pragma unroll 4
  for (int k = 0; k < CCH; k += 4) {
    const int kb = k + 2 * half;
    v2f a;
    a.x = A[l * CCH + kb]; a.y = A[l * CCH + kb + 1];
    const float* Br = B + (size_t)kb * HWPX;
#pragma unroll
    for (int pt = 0; pt < 4; ++pt) {
      v2f b;
      b.x = Br[pt * 16 + l]; b.y = Br[HWPX + pt * 16 + l];
      acc[pt] = WMMA_F32(a, b, acc[pt]);
    }
  }
  float* out = corr + (size_t)((br * NIMG + n) * NIMG) * HWPX + p0;
#pragma unroll
  for (int pt = 0; pt < 4; ++pt) {
    const int p = pt * 16 + l;
    const float s = invn[n * HWPX + p0 + p];
#pragma unroll
    for (int r = 0; r < 8; ++r) out[(size_t)(r + 8 * half) * HWPX + p] = acc[pt][r] * s;
  }
}

// 1/||corr row||; grid 512 x 256
__global__ void rownorm_kernel(const float* __restrict__ corr, float* __restrict__ rn) {
  int row = blockIdx.x, tid = threadIdx.x;
  const float* r = corr + (size_t)row * HWPX;
  float s = 0.0f;
  for (int p = tid; p < HWPX; p += 256) { float v = r[p]; s += v * v; }
  __shared__ float red[256];
  red[tid] = s; __syncthreads();
  for (int off = 128; off; off >>= 1) { if (tid < off) red[tid] += red[tid + off]; __syncthreads(); }
  if (tid == 0) rn[row] = 1.0f / fmaxf(sqrtf(red[0]), EPSF);
}

// att[br][n][p] = minmax( sum_m corr[br][n][m][p]*rn[br][n][m] ); grid 32 x 256
__global__ void att_kernel(const float* __restrict__ corr, const float* __restrict__ rn,
                           float* __restrict__ att) {
  int bn = blockIdx.x, tid = threadIdx.x;               // bn = br*16+n
  const float* cb = corr + (size_t)bn * NIMG * HWPX;
  const float* rb = rn + bn * NIMG;
  float vals[16], mn = 3.4e38f, mx = -3.4e38f;
#pragma unroll
  for (int i = 0; i < 16; ++i) {
    int p = tid + i * 256;
    float s = 0.0f;
    for (int m = 0; m < NIMG; ++m) s += cb[(size_t)m * HWPX + p] * rb[m];
    vals[i] = s; mn = fminf(mn, s); mx = fmaxf(mx, s);
  }
  __shared__ float rmn[256], rmx[256];
  rmn[tid] = mn; rmx[tid] = mx; __syncthreads();
  for (int off = 128; off; off >>= 1) {
    if (tid < off) { rmn[tid] = fminf(rmn[tid], rmn[tid + off]); rmx[tid] = fmaxf(rmx[tid], rmx[tid + off]); }
    __syncthreads();
  }
  float MN = rmn[0], inv = 1.0f / (rmx[0] - rmn[0] + EPSF);
#pragma unroll
  for (int i = 0; i < 16; ++i) att[(size_t)bn * HWPX + tid + i * 256] = (vals[i] - MN) * inv;
}

// score[br][g] = NFs(n,:,p) . r_co_proxy[br]; grid 256 x 256
__global__ void score_kernel(const float* __restrict__ feats, const float* __restrict__ invn,
                             const float* __restrict__ rproxy, float* __restrict__ score) {
  int g = blockIdx.x * 256 + threadIdx.x;
  int n = g >> 12, p = g & (HWPX - 1);
  const float* f = feats + (size_t)n * CHW + p;
  float sf = 0.0f, sb = 0.0f;
  for (int c = 0; c < CCH; ++c) {
    float v = f[(size_t)c * HWPX];
    sf += v * rproxy[c]; sb += v * rproxy[CCH + c];
  }
  float iv = invn[g];
  score[g] = sf * iv; score[NPIX + g] = sb * iv;
}

__device__ __forceinline__ unsigned int fmap(float f) {
  unsigned int u = __float_as_uint(f);
  return (u & 0x80000000u) ? ~u : (u | 0x80000000u);   // order-preserving map
}

// exact top-32, descending score, ties -> lower index (JAX top_k semantics); grid 2 x 1024
__global__ void topk_kernel(const float* __restrict__ score, int* __restrict__ idx) {
  int br = blockIdx.x, tid = threadIdx.x;
  const float* s = score + (size_t)br * NPIX;
  __shared__ unsigned long long red[1024];
  unsigned long long prev = 0xFFFFFFFFFFFFFFFFull;
  for (int it = 0; it < 32; ++it) {
    unsigned long long best = 0ull;
    for (int g = tid; g < NPIX; g += 1024) {
      unsigned long long k = ((unsigned long long)fmap(s[g]) << 32)
                           | (unsigned long long)(0xFFFFFFFFu - (unsigned)g);
      if (k < prev && k > best) best = k;
    }
    red[tid] = best; __syncthreads();
    for (int off = 512; off; off >>= 1) {
      if (tid < off && red[tid + off] > red[tid]) red[tid] = red[tid + off];
      __syncthreads();
    }
    unsigned long long sel = red[0];
    __syncthreads();
    if (tid == 0) idx[br * 32 + it] = (int)(0xFFFFFFFFu - (unsigned)(sel & 0xFFFFFFFFull));
    prev = sel;
  }
}

// co_rep[br][k][c] = NFs at selected pixel; grid (32,2) x 256
__global__ void gather_kernel(const float* __restrict__ feats, const float* __restrict__ invn,
                              const int* __restrict__ idx, float* __restrict__ corep) {
  int k = blockIdx.x, br = blockIdx.y, c = threadIdx.x;
  int g = idx[br * 32 + k]; int n = g >> 12, p = g & (HWPX - 1);
  corep[(br * 32 + k) * CCH + c] = feats[(size_t)n * CHW + (size_t)c * HWPX + p] * invn[g];
}

// tatt[br][n][k][p] = att*invn * sum_c co_rep[br][k][c]*feats[n][c][p]
// wave: 32 rows x 32 pixels (2x2 register block); grid (32,16,2) x 128
__global__ void cofs_gemm(const float* __restrict__ feats, const float* __restrict__ invn,
                          const float* __restrict__ att, const float* __restrict__ corep,
                          float* __restrict__ tatt) {
  const int lane = threadIdx.x & 31, wid = threadIdx.x >> 5;
  const int half = lane >> 4, l = lane & 15;
  const int p0 = (blockIdx.x * 4 + wid) << 5;                // 2 pixel tiles
  const int n = blockIdx.y, br = blockIdx.z;
  const float* A = corep + br * 32 * CCH;                    // [32][256]
  const float* B = feats + (size_t)n * CHW + p0;
  v8f acc[2][2] = {};
#pragma unroll 4
  for (int k = 0; k < CCH; k += 4) {
    const int kb = k + 2 * half;
    const float* Br = B + (size_t)kb * HWPX;
    v2f a0, a1, b0, b1;
    a0.x = A[l * CCH + kb];        a0.y = A[l * CCH + kb + 1];
    a1.x = A[(16 + l) * CCH + kb]; a1.y = A[(16 + l) * CCH + kb + 1];
    b0.x = Br[l];                  b0.y = Br[HWPX + l];
    b1.x = Br[16 + l];             b1.y = Br[HWPX + 16 + l];
    acc[0][0] = WMMA_F32(a0, b0, acc[0][0]);
    acc[0][1] = WMMA_F32(a0, b1, acc[0][1]);
    acc[1][0] = WMMA_F32(a1, b0, acc[1][0]);
    acc[1][1] = WMMA_F32(a1, b1, acc[1][1]);
  }
  float* out = tatt + (size_t)((br * NIMG + n) * 32) * HWPX + p0;
#pragma unroll
  for (int pt = 0; pt < 2; ++pt) {
    const int p = pt * 16 + l;
    const float sc = att[(size_t)(br * NIMG + n) * HWPX + p0 + p] * invn[n * HWPX + p0 + p];
#pragma unroll
    for (int r = 0; r < 8; ++r) {
      out[(size_t)(r + 8 * half) * HWPX + p]      = acc[0][pt][r] * sc;
      out[(size_t)(16 + r + 8 * half) * HWPX + p] = acc[1][pt][r] * sc;
    }
  }
}

// fore/back[n][o][p] = sum_k w_sub[o][k]*tatt[br][n][k][p] + b_sub[o]
// wave: 64 rows x 32 pixels (4x2 block); grid (32,4,32) x 128
__global__ void sub_out_gemm(const float* __restrict__ tatt, const float* __restrict__ wsub,
                             const float* __restrict__ bsub, float* __restrict__ forebase) {
  const int lane = threadIdx.x & 31, wid = threadIdx.x >> 5;
  const int half = lane >> 4, l = lane & 15;
  const int p0 = (blockIdx.x * 4 + wid) << 5;                // 2 pixel tiles
  const int m0 = blockIdx.y << 6;                            // 4 m-tiles
  const int bn = blockIdx.z;                                 // br*16+n
  const float* A = wsub + m0 * 32;                           // ldA = 32
  const float* B = tatt + (size_t)(bn * 32) * HWPX + p0;
  v8f acc[4][2] = {};
#pragma unroll 2
  for (int k = 0; k < 32; k += 4) {
    const int kb = k + 2 * half;
    const float* Br = B + (size_t)kb * HWPX;
    v2f b0, b1;
    b0.x = Br[l];      b0.y = Br[HWPX + l];
    b1.x = Br[16 + l]; b1.y = Br[HWPX + 16 + l];
#pragma unroll
    for (int t = 0; t < 4; ++t) {
      v2f a;
      a.x = A[(t * 16 + l) * 32 + kb]; a.y = A[(t * 16 + l) * 32 + kb + 1];
      acc[t][0] = WMMA_F32(a, b0, acc[t][0]);
      acc[t][1] = WMMA_F32(a, b1, acc[t][1]);
    }
  }
  float* out = forebase + (size_t)bn * CHW + p0;
#pragma unroll
  for (int t = 0; t < 4; ++t)
#pragma unroll
    for (int pt = 0; pt < 2; ++pt)
#pragma unroll
      for (int r = 0; r < 8; ++r) {
        const int m = m0 + t * 16 + r + 8 * half;
        out[(size_t)m * HWPX + pt * 16 + l] = acc[t][pt][r] + bsub[m];
      }
}

// enhanced[n][o][p] = sum_{c<512} cat[c][p]*w_out[o][c] + b_out[o]
// wave: 64 rows x 32 pixels (4x2 block); grid (32,4,16) x 128
__global__ void fuse_gemm(const float* __restrict__ forebase, const float* __restrict__ wout,
                          const float* __restrict__ bout, float* __restrict__ enh) {
  const int lane = threadIdx.x & 31, wid = threadIdx.x >> 5;
  const int half = lane >> 4, l = lane & 15;
  const int p0 = (blockIdx.x * 4 + wid) << 5;                // 2 pixel tiles
  const int m0 = blockIdx.y << 6;                            // 4 m-tiles
  const int n = blockIdx.z;
  const float* A = wout + m0 * 512;                          // ldA = 512
  v8f acc[4][2] = {};
#pragma unroll 2
  for (int k = 0; k < 512; k += 4) {
    const int kk = (k < CCH) ? k : (k - CCH);
    const int kb = kk + 2 * half;
    const float* Br = forebase + (size_t)((k < CCH) ? 0 : NIMG * CHW)
                    + (size_t)n * CHW + (size_t)kb * HWPX + p0;
    v2f b0, b1;
    b0.x = Br[l];      b0.y = Br[HWPX + l];
    b1.x = Br[16 + l]; b1.y = Br[HWPX + 16 + l];
#pragma unroll
    for (int t = 0; t < 4; ++t) {
      v2f a;
      a.x = A[(t * 16 + l) * 512 + k + 2 * half];
      a.y = A[(t * 16 + l) * 512 + k + 2 * half + 1];
      acc[t][0] = WMMA_F32(a, b0, acc[t][0]);
      acc[t][1] = WMMA_F32(a, b1, acc[t][1]);
    }
  }
  float* out = enh + (size_t)n * CHW + p0;
#pragma unroll
  for (int t = 0; t < 4; ++t)
#pragma unroll
    for (int pt = 0; pt < 2; ++pt)
#pragma unroll
      for (int r = 0; r < 8; ++r) {
        const int m = m0 + t * 16 + r + 8 * half;
        // enhanced is never re-read on-device: stream it past L2 (NT hint)
        __builtin_nontemporal_store(acc[t][pt][r] + bout[m],
                                    out + (size_t)m * HWPX + pt * 16 + l);
      }
}

// =============== launch ===============
extern "C" void kernel_launch(void* const* d_in, const int* in_sizes, int n_in,
                              void* d_out, int out_size, void* d_ws, size_t ws_size,
                              hipStream_t stream) {
  const float* feats = (const float*)d_in[0];
  const float* sisms = (const float*)d_in[1];
  const float* wsub  = (const float*)d_in[2];
  const float* bsub  = (const float*)d_in[3];
  const float* wout  = (const float*)d_in[4];
  const float* bout  = (const float*)d_in[5];
  float* out = (float*)d_out;
  float* ws  = (float*)d_ws;

  float* sig    = ws + WS_SIG;
  float* invn   = ws + WS_INVN;
  float* wei    = ws + WS_WEI;
  float* proxy  = ws + WS_PROXY;
  float* rproxy = ws + WS_RPROXY;
  float* rn     = ws + WS_RN;
  float* att    = ws + WS_ATT;
  float* score  = ws + WS_SCORE;
  float* corep  = ws + WS_COREP;
  int*   idx    = (int*)(ws + WS_IDX);
  float* corr   = ws + WS_CORR;
  float* tatt   = ws + WS_TATT;

  float* enh      = out;                       // [16][256][4096]
  float* forebase = out + (size_t)NIMG * CHW;  // fore, then back

  prep_kernel    <<<dim3(NPIX / 256), 256, 0, stream>>>(feats, sisms, sig, invn);
  weighted_kernel<<<dim3(NIMG * CCH), 256, 0, stream>>>(feats, sig, invn, wei);
  proxy_kernel   <<<dim3(17, 2), 256, 0, stream>>>(wei, proxy, rproxy);
  corr_gemm      <<<dim3(16, NIMG, 2), 128, 0, stream>>>(feats, invn, proxy, corr);
  rownorm_kernel <<<dim3(2 * NIMG * NIMG), 256, 0, stream>>>(corr, rn);
  att_kernel     <<<dim3(2 * NIMG), 256, 0, stream>>>(corr, rn, att);
  score_kernel   <<<dim3(NPIX / 256), 256, 0, stream>>>(feats, invn, rproxy, score);
  topk_kernel    <<<dim3(2), 1024, 0, stream>>>(score, idx);
  gather_kernel  <<<dim3(32, 2), 256, 0, stream>>>(feats, invn, idx, corep);
  cofs_gemm      <<<dim3(32, NIMG, 2), 128, 0, stream>>>(feats, invn, att, corep, tatt);
  sub_out_gemm   <<<dim3(32, 4, 2 * NIMG), 128, 0, stream>>>(tatt, wsub, bsub, forebase);
  fuse_gemm      <<<dim3(32, 4, NIMG), 128, 0, stream>>>(forebase, wout, bout, enh);
}